// SwinTransformerBlockSequence_49589692399804
// MI455X (gfx1250) — compile-verified
//
#include <hip/hip_runtime.h>
#include <math.h>

// ---------------- constants ----------------
namespace {
constexpr int kB    = 8;      // batch
constexpr int kS    = 4;      // sequence
constexpr int kWS   = 7;      // window
constexpr int kSh   = 3;      // shift
constexpr int kDim  = 96;
constexpr int kNH   = 3;
constexpr int kHD   = 32;
constexpr int kN    = 196;    // tokens / window
constexpr int kH    = 56, kW = 56;
constexpr int kL    = kS * kH * kW;        // 12544
constexpr int kBW   = kB * 64;             // 512 windows total
constexpr int kRows = kBW * kN;            // 100352 token-rows
constexpr int kHid  = 384;
constexpr float kScale = 0.17677669529663687f; // 32^-0.5
}

typedef float v2f __attribute__((ext_vector_type(2)));
typedef float v8f __attribute__((ext_vector_type(8)));

static __device__ __forceinline__ v8f wmma4(v2f a, v2f b, v8f c) {
#if defined(__HIP_DEVICE_COMPILE__)
  // D = A(16x4,f32) * B(4x16,f32) + C(16x16,f32)  -- fp32 matrix core
  return __builtin_amdgcn_wmma_f32_16x16x4_f32(false, a, false, b, (short)0, c,
                                               false, false);
#else
  (void)a; (void)b; return c;
#endif
}

static __device__ __forceinline__ float redmax16(float v) {
  v = fmaxf(v, __shfl_xor(v, 1, 32));
  v = fmaxf(v, __shfl_xor(v, 2, 32));
  v = fmaxf(v, __shfl_xor(v, 4, 32));
  v = fmaxf(v, __shfl_xor(v, 8, 32));
  return v;
}
static __device__ __forceinline__ float redsum16(float v) {
  v += __shfl_xor(v, 1, 32);
  v += __shfl_xor(v, 2, 32);
  v += __shfl_xor(v, 4, 32);
  v += __shfl_xor(v, 8, 32);
  return v;
}
static __device__ __forceinline__ float redsum32(float v) {
  v += __shfl_xor(v, 1, 32);
  v += __shfl_xor(v, 2, 32);
  v += __shfl_xor(v, 4, 32);
  v += __shfl_xor(v, 8, 32);
  v += __shfl_xor(v, 16, 32);
  return v;
}
static __device__ __forceinline__ int regc(int c) {
  return c < (kH - kWS) ? 0 : (c < (kH - kSh) ? 1 : 2);  // <49, <53, else
}
static __device__ __forceinline__ void lds_wait() {
#if defined(__HIP_DEVICE_COMPILE__)
  asm volatile("s_wait_dscnt 0x0" ::: "memory");
#endif
}

// ============ K1: LN1 + roll(-3,-3) + window partition ============
// one wave per token; 96 channels = 3 per lane
__global__ __launch_bounds__(256) void k_ln1_part(const float* __restrict__ x,
                                                  const float* __restrict__ w,
                                                  const float* __restrict__ b,
                                                  float* __restrict__ hwin) {
  const int wave = threadIdx.x >> 5, lane = threadIdx.x & 31;
  const int tk = blockIdx.x * 8 + wave;                  // 0..100351
  const int b_ = tk / kN, t = tk % kN;
  const int bb = b_ >> 6, wi = b_ & 63;
  const int wr = wi >> 3, wc = wi & 7;
  const int s = t / 49, p = t % 49;
  const int wh = p / 7, ww = p % 7;
  const int ys = (wr * 7 + wh + kSh) % kH;               // roll -3
  const int xs = (wc * 7 + ww + kSh) % kW;
  const long l = (long)s * (kH * kW) + ys * kW + xs;
  const float* row = x + ((long)bb * kL + l) * kDim;
  float v0 = row[lane], v1 = row[lane + 32], v2 = row[lane + 64];
  float mu = redsum32(v0 + v1 + v2) * (1.0f / 96.0f);
  float d0 = v0 - mu, d1 = v1 - mu, d2 = v2 - mu;
  float var = redsum32(d0 * d0 + d1 * d1 + d2 * d2) * (1.0f / 96.0f);
  float rstd = rsqrtf(var + 1e-5f);
  float* out = hwin + (long)tk * kDim;
  out[lane]      = d0 * rstd * w[lane]      + b[lane];
  out[lane + 32] = d1 * rstd * w[lane + 32] + b[lane + 32];
  out[lane + 64] = d2 * rstd * w[lane + 64] + b[lane + 64];
}

// ============ K2: relative position bias matrix (NH,196,196) ============
__global__ void k_bias(const float* __restrict__ rpb, float* __restrict__ bt) {
  int idx = blockIdx.x * 256 + threadIdx.x;
  if (idx >= kN * kN) return;
  int i = idx / kN, j = idx % kN;
  // reference flat order: t = ch*(WS*S) + cw*S + cs
  int chi = i / 28, cwi = (i / 4) % 7, csi = i & 3;
  int chj = j / 28, cwj = (j / 4) % 7, csj = j & 3;
  int r = (chi - chj + 6) + (cwi - cwj + 6) * 13 + (csi - csj + 3) * 169;
  for (int h = 0; h < kNH; h++)
    bt[(long)h * kN * kN + idx] = rpb[r * kNH + h];
}

// ============ K3: QKV GEMM  (100352x96)@(96x288)^T + bias ============
// one wave per 16x16 output tile; weight col-tile staged in LDS
__global__ __launch_bounds__(256) void k_qkv(const float* __restrict__ hwin,
                                             const float* __restrict__ wq,
                                             const float* __restrict__ bq,
                                             float* __restrict__ q,
                                             float* __restrict__ k,
                                             float* __restrict__ v) {
  __shared__ float wsm[16 * kDim];
  const int ntile = blockIdx.y;
  for (int e = threadIdx.x; e < 16 * kDim; e += 256) {
    int col = ntile * 16 + e / kDim;
    wsm[e] = wq[col * kDim + (e % kDim)];
  }
  __syncthreads();
  const int wave = threadIdx.x >> 5, lane = threadIdx.x & 31;
  const int half = lane >> 4, ln = lane & 15;
  const int mtile = blockIdx.x * 8 + wave;
  const int row = mtile * 16 + ln;
  const int col = ntile * 16 + ln;
  v8f acc;
  float bv = bq[col];
  for (int r = 0; r < 8; r++) acc[r] = bv;
  const float* arow = hwin + (long)row * kDim;
  for (int k0 = 0; k0 < kDim; k0 += 4) {
    int kk = k0 + half * 2;
    v2f a = {arow[kk], arow[kk + 1]};
    v2f bf = {wsm[ln * kDim + kk], wsm[ln * kDim + kk + 1]};
    acc = wmma4(a, bf, acc);
  }
  const int which = col / kDim, h = (col % kDim) / kHD, hd = col % kHD;
  float* dst = which == 0 ? q : (which == 1 ? k : v);
  for (int r = 0; r < 8; r++) {
    int grow = mtile * 16 + r + 8 * half;
    int bw = grow / kN, n = grow % kN;
    dst[(((long)bw * kNH + h) * kN + n) * kHD + hd] = acc[r];
  }
}

// ============ K4: fused flash attention per (window, head) ============
__global__ __launch_bounds__(256) void k_attn(const float* __restrict__ q,
                                              const float* __restrict__ kmat,
                                              const float* __restrict__ vmat,
                                              const float* __restrict__ bt,
                                              float* __restrict__ out) {
  __shared__ float ks[208 * kHD];   // K padded, zero-filled
  __shared__ float vs[208 * kHD];   // V padded
  __shared__ float stage[8][256];   // per-wave C->A transpose stage
  const int b_ = blockIdx.x / kNH, h = blockIdx.x % kNH;
  const long base = ((long)b_ * kNH + h) * kN * kHD;
  for (int e = threadIdx.x; e < 208 * kHD; e += 256) {
    int n = e >> 5;
    float kv = 0.f, vv = 0.f;
    if (n < kN) { kv = kmat[base + e]; vv = vmat[base + e]; }
    ks[e] = kv; vs[e] = vv;
  }
  __syncthreads();
  const int wave = threadIdx.x >> 5, lane = threadIdx.x & 31;
  const int half = lane >> 4, ln = lane & 15;
  const int wi = b_ & 63, wr = wi >> 3, wc = wi & 7;
  float* st = stage[wave];
  const float* bh = bt + (long)h * kN * kN;

  for (int qt = wave; qt < 13; qt += 8) {               // 13 query tiles (196=12*16+4)
    // ---- Q fragment (pre-scaled), 16 regs: half-row per lane ----
    float ql[16];
    const int qi0 = qt * 16 + ln;
    for (int s8 = 0; s8 < 8; s8++) {
      float a0 = 0.f, a1 = 0.f;
      if (qi0 < kN) {
        const float* qr = q + base + (long)qi0 * kHD + s8 * 4 + half * 2;
        a0 = qr[0]; a1 = qr[1];
      }
      ql[2 * s8] = a0 * kScale; ql[2 * s8 + 1] = a1 * kScale;
    }
    // per-lane row stats (rows r+8*half of this tile)
    float rmax[8], rsum[8];
    v8f o0, o1;
    int cqr[8]; bool qv[8];
    for (int r = 0; r < 8; r++) {
      rmax[r] = -1e30f; rsum[r] = 0.f; o0[r] = 0.f; o1[r] = 0.f;
      int qi = qt * 16 + r + 8 * half;
      qv[r] = qi < kN;
      if (qv[r]) {
        int pq = qi % 49;
        cqr[r] = regc(wr * 7 + pq / 7) * 3 + regc(wc * 7 + pq % 7);
      } else cqr[r] = 0;
    }
    for (int kt = 0; kt < 13; kt++) {
      // ---- S tile = (q*scale) @ K^T via 8 fp32 WMMAs ----
      v8f sacc; for (int r = 0; r < 8; r++) sacc[r] = 0.f;
      for (int s8 = 0; s8 < 8; s8++) {
        int kk = s8 * 4 + half * 2;
        int n = kt * 16 + ln;
        v2f a = {ql[2 * s8], ql[2 * s8 + 1]};
        v2f bf = {ks[n * kHD + kk], ks[n * kHD + kk + 1]};
        sacc = wmma4(a, bf, sacc);
      }
      // ---- + bias + shift-mask (computed inline) ----
      const int j = kt * 16 + ln;
      const bool jvld = j < kN;
      int cj = 0;
      if (jvld) {
        int pj = j % 49;
        cj = regc(wr * 7 + pj / 7) * 3 + regc(wc * 7 + pj % 7);
      }
      float sv[8];
      for (int r = 0; r < 8; r++) {
        float val = -1e30f;
        if (jvld && qv[r]) {
          int qi = qt * 16 + r + 8 * half;
          val = sacc[r] + bh[qi * kN + j] + (cqr[r] == cj ? 0.f : -100.f);
        }
        sv[r] = val;
      }
      // ---- online softmax update ----
      for (int r = 0; r < 8; r++) {
        float tm = redmax16(sv[r]);
        float nm = fmaxf(rmax[r], tm);
        float f = __expf(rmax[r] - nm);
        float pe = __expf(sv[r] - nm);
        rsum[r] = rsum[r] * f + redsum16(pe);
        rmax[r] = nm;
        o0[r] *= f; o1[r] *= f;
        st[(r + 8 * half) * 16 + ln] = pe;     // stage P (C-layout write)
      }
      lds_wait();                              // same-wave DS RAW
      // ---- O += P @ V  (read P back in A-layout) ----
      for (int s4 = 0; s4 < 4; s4++) {
        int kk = s4 * 4 + half * 2;
        int kidx = kt * 16 + kk;
        v2f a = {st[ln * 16 + kk], st[ln * 16 + kk + 1]};
        v2f bl = {vs[kidx * kHD + ln], vs[(kidx + 1) * kHD + ln]};
        v2f bhh = {vs[kidx * kHD + 16 + ln], vs[(kidx + 1) * kHD + 16 + ln]};
        o0 = wmma4(a, bl, o0);
        o1 = wmma4(a, bhh, o1);
      }
    }
    // ---- normalize + store (b_, n, h*32+hd) ----
    for (int r = 0; r < 8; r++) {
      int qi = qt * 16 + r + 8 * half;
      if (qi < kN) {
        float inv = 1.f / rsum[r];
        float* orow = out + ((long)b_ * kN + qi) * kDim + h * kHD;
        orow[ln] = o0[r] * inv;
        orow[16 + ln] = o1[r] * inv;
      }
    }
  }
}

// ============ K5: proj GEMM + window reverse + roll(+3) + residual ============
__global__ __launch_bounds__(256) void k_proj(const float* __restrict__ ain,
                                              const float* __restrict__ wp,
                                              const float* __restrict__ bp,
                                              const float* __restrict__ xin,
                                              float* __restrict__ x1) {
  __shared__ float wsm[16 * kDim];
  const int ntile = blockIdx.y;
  for (int e = threadIdx.x; e < 16 * kDim; e += 256) {
    int col = ntile * 16 + e / kDim;
    wsm[e] = wp[col * kDim + (e % kDim)];
  }
  __syncthreads();
  const int wave = threadIdx.x >> 5, lane = threadIdx.x & 31;
  const int half = lane >> 4, ln = lane & 15;
  const int mtile = blockIdx.x * 8 + wave;
  const int row = mtile * 16 + ln;
  const int col = ntile * 16 + ln;
  v8f acc;
  float bv = bp[col];
  for (int r = 0; r < 8; r++) acc[r] = bv;
  const float* arow = ain + (long)row * kDim;
  for (int k0 = 0; k0 < kDim; k0 += 4) {
    int kk = k0 + half * 2;
    v2f a = {arow[kk], arow[kk + 1]};
    v2f bf = {wsm[ln * kDim + kk], wsm[ln * kDim + kk + 1]};
    acc = wmma4(a, bf, acc);
  }
  for (int r = 0; r < 8; r++) {
    int grow = mtile * 16 + r + 8 * half;
    int b_ = grow / kN, t = grow % kN;
    int bb = b_ >> 6, wi = b_ & 63;
    int wr = wi >> 3, wc = wi & 7;
    int s = t / 49, p = t % 49;
    int y = (wr * 7 + p / 7 + kSh) % kH;                 // reverse roll +3
    int xx = (wc * 7 + p % 7 + kSh) % kW;
    long o = ((long)bb * kL + (long)s * (kH * kW) + y * kW + xx) * kDim + col;
    x1[o] = xin[o] + acc[r];
  }
}

// ============ K6: fused LN2 + fc1 + exact GELU + fc2 + residual ============
__global__ __launch_bounds__(256) void k_mlp(const float* __restrict__ x1,
                                             const float* __restrict__ n2w,
                                             const float* __restrict__ n2b,
                                             const float* __restrict__ w1,
                                             const float* __restrict__ b1,
                                             const float* __restrict__ w2,
                                             const float* __restrict__ b2,
                                             float* __restrict__ out) {
  __shared__ float xs[16 * kDim];    // raw rows (residual)
  __shared__ float lns[16 * kDim];   // LN2 rows
  __shared__ float hs[16 * kHid];    // GELU(fc1) hidden tile
  const long rowbase = (long)blockIdx.x * 16;
  for (int e = threadIdx.x; e < 16 * kDim; e += 256)
    xs[e] = x1[rowbase * kDim + e];
  __syncthreads();
  const int wave = threadIdx.x >> 5, lane = threadIdx.x & 31;
  for (int m = wave; m < 16; m += 8) {
    const float* rr = xs + m * kDim;
    float v0 = rr[lane], v1 = rr[lane + 32], v2 = rr[lane + 64];
    float mu = redsum32(v0 + v1 + v2) * (1.0f / 96.0f);
    float d0 = v0 - mu, d1 = v1 - mu, d2 = v2 - mu;
    float var = redsum32(d0 * d0 + d1 * d1 + d2 * d2) * (1.0f / 96.0f);
    float rstd = rsqrtf(var + 1e-5f);
    lns[m * kDim + lane]      = d0 * rstd * n2w[lane]      + n2b[lane];
    lns[m * kDim + lane + 32] = d1 * rstd * n2w[lane + 32] + n2b[lane + 32];
    lns[m * kDim + lane + 64] = d2 * rstd * n2w[lane + 64] + n2b[lane + 64];
  }
  __syncthreads();
  const int half = lane >> 4, ln = lane & 15;
  // fc1: 24 col-tiles over 8 waves
  for (int i = 0; i < 3; i++) {
    int col = (wave + 8 * i) * 16 + ln;
    v8f acc;
    float bv = b1[col];
    for (int r = 0; r < 8; r++) acc[r] = bv;
    for (int k0 = 0; k0 < kDim; k0 += 4) {
      int kk = k0 + half * 2;
      v2f a = {lns[ln * kDim + kk], lns[ln * kDim + kk + 1]};
      v2f bf = {w1[col * kDim + kk], w1[col * kDim + kk + 1]};
      acc = wmma4(a, bf, acc);
    }
    for (int r = 0; r < 8; r++) {
      float vv = acc[r];
      float g = 0.5f * vv * (1.f + erff(vv * 0.70710678118654752f)); // exact GELU
      hs[(r + 8 * half) * kHid + col] = g;
    }
  }
  __syncthreads();
  // fc2: 6 col-tiles, waves 0..5
  if (wave < 6) {
    int col = wave * 16 + ln;
    v8f acc;
    float bv = b2[col];
    for (int r = 0; r < 8; r++) acc[r] = bv;
    for (int k0 = 0; k0 < kHid; k0 += 4) {
      int kk = k0 + half * 2;
      v2f a = {hs[ln * kHid + kk], hs[ln * kHid + kk + 1]};
      v2f bf = {w2[col * kHid + kk], w2[col * kHid + kk + 1]};
      acc = wmma4(a, bf, acc);
    }
    for (int r = 0; r < 8; r++) {
      int m = r + 8 * half;
      out[(rowbase + m) * kDim + col] = acc[r] + xs[m * kDim + col];
    }
  }
}

// ============================== launch ==============================
extern "C" void kernel_launch(void* const* d_in, const int* in_sizes, int n_in,
                              void* d_out, int out_size, void* d_ws, size_t ws_size,
                              hipStream_t stream) {
  const float* x      = (const float*)d_in[0];
  const float* n1w    = (const float*)d_in[1];
  const float* n1b    = (const float*)d_in[2];
  const float* qkv_w  = (const float*)d_in[3];
  const float* qkv_b  = (const float*)d_in[4];
  const float* rpb    = (const float*)d_in[5];
  const float* proj_w = (const float*)d_in[6];
  const float* proj_b = (const float*)d_in[7];
  const float* n2w    = (const float*)d_in[8];
  const float* n2b    = (const float*)d_in[9];
  const float* fc1_w  = (const float*)d_in[10];
  const float* fc1_b  = (const float*)d_in[11];
  const float* fc2_w  = (const float*)d_in[12];
  const float* fc2_b  = (const float*)d_in[13];
  float* outp = (float*)d_out;

  // workspace layout (floats); peak ~193 MB
  const size_t SZ = (size_t)kRows * kDim;          // 9,633,792
  float* ws    = (float*)d_ws;
  float* hwin  = ws;                               // also reused as attn output
  float* qb    = ws + SZ;
  float* kb    = ws + 2 * SZ;
  float* vb    = ws + 3 * SZ;
  float* x1    = ws + 4 * SZ;
  float* biasT = ws + 5 * SZ;                      // 3*196*196 floats

  // K1: LN1 + shift + partition
  k_ln1_part<<<kRows / 8, 256, 0, stream>>>(x, n1w, n1b, hwin);
  // K2: relative-position bias matrix
  k_bias<<<(kN * kN + 255) / 256, 256, 0, stream>>>(rpb, biasT);
  // K3: QKV GEMM
  k_qkv<<<dim3(kRows / 16 / 8, (3 * kDim) / 16), 256, 0, stream>>>(
      hwin, qkv_w, qkv_b, qb, kb, vb);
  // K4: fused flash attention (overwrites hwin with attention output)
  k_attn<<<kBW * kNH, 256, 0, stream>>>(qb, kb, vb, biasT, hwin);
  // K5: proj + window reverse + unshift + residual
  k_proj<<<dim3(kRows / 16 / 8, kDim / 16), 256, 0, stream>>>(
      hwin, proj_w, proj_b, x, x1);
  // K6: fused LN2 + MLP + residual -> d_out
  k_mlp<<<kRows / 16, 256, 0, stream>>>(x1, n2w, n2b, fc1_w, fc1_b, fc2_w,
                                        fc2_b, outp);
}